// PatchNeighborSearcher_23708219474115
// MI455X (gfx1250) — compile-verified
//
#include <hip/hip_runtime.h>
#include <stdint.h>

// PatchNeighborSearcher for MI455X (gfx1250).
// Shapes (from reference): B=8, H=W=16, L=64, C=64.
//   out[b, h*W+w, l*8+n, c] = E[b, (h+dy[n])*W + (w+dx[n]), l, c]  (0 if OOB)
//
// Pure data movement: 32 MB input (L2-resident, reused x8) + 256 MB streamed
// output. HBM floor @ 23.3 TB/s ~= 12.4 us. Strategy:
//   * one WG (256 thr = 8 waves) per (b,p) output tile (CONTIGUOUS 128 KB)
//   * wave n async-copies neighbor n's contiguous 16 KB block into LDS
//     (global_load_async_to_lds_b128, saddr form), OOB slots zero-filled
//   * all threads then stream the tile out in linear float4 order with
//     NON-TEMPORAL stores so the 256 MB write stream does not evict the
//     32 MB input from L2 (which is what enables the x8 read reuse).

namespace {

typedef float v4f __attribute__((ext_vector_type(4)));

constexpr int H = 16, W = 16, L = 64, C = 64;
constexpr int P       = H * W;           // 256 patches
constexpr int BLK_F4  = L * (C / 4);     // 1024 v4f = 16 KB per input block
constexpr int TILE_F4 = 8 * BLK_F4;      // 8192 v4f = 128 KB per output tile

__global__ __launch_bounds__(256)
void patch_neighbor_kernel(const float* __restrict__ E, v4f* __restrict__ out)
{
    __shared__ v4f smem[8 * BLK_F4];     // 128 KB: slot n = neighbor n's block

    const int tile = blockIdx.x;         // b*P + p
    const int p    = tile & (P - 1);
    const int h    = p >> 4;             // p / W
    const int w    = p & (W - 1);        // p % W

    const int tid  = threadIdx.x;
    const int wave = tid >> 5;           // 8 waves: wave n stages neighbor n
    const int lane = tid & 31;

    // ---- Phase 1: async-stage eight neighbor blocks into LDS --------------
    // neighbor n -> 3x3 cell skipping center: (dy,dx) in row-major order
    const int nid = (wave < 4) ? wave : wave + 1;
    const int dy  = nid / 3 - 1;
    const int dx  = nid % 3 - 1;
    const int hn  = h + dy;
    const int wn  = w + dx;
    const bool inb = (hn >= 0) & (hn < H) & (wn >= 0) & (wn < W);

    if (inb) {
        // (tile - p) == b*P ; neighbor block is contiguous 16 KB at nb_tile*L*C
        const int nb_tile = (tile - p) + hn * W + wn;
        const uint64_t gbase =
            (uint64_t)(uintptr_t)(E + (size_t)nb_tile * (L * C));  // wave-uniform
        // low 32 bits of the flat shared address == LDS byte offset
        const uint32_t lds_slot = (uint32_t)(uintptr_t)(&smem[wave * BLK_F4]);
        #pragma unroll
        for (int j = 0; j < 32; ++j) {
            const uint32_t voff = (uint32_t)((j * 32 + lane) * 16); // bytes
            const uint32_t ldsa = lds_slot + voff;
            // gfx1250 async copy: copies 16 B per lane from global address gbase + voff to LDS address ldsa
            asm volatile("global_load_async_to_lds_b128 %0, %1, %2"
                         :
                         : "v"(ldsa), "v"(voff), "s"(gbase)
                         : "memory");
        }
        asm volatile("s_wait_asynccnt 0" ::: "memory");
    } else {
        // zero-pad: fill this neighbor's LDS slot with zeros (ds_store_b128)
        const v4f z = (v4f)0.0f;
        #pragma unroll
        for (int j = 0; j < 32; ++j)
            smem[wave * BLK_F4 + j * 32 + lane] = z;
    }

    __syncthreads();

    // ---- Phase 2: stream the 128 KB tile out, fully coalesced, TH=NT ------
    // out tile is contiguous: o4 = (l*8+n)*16 + c4, v4f units.
    v4f* __restrict__ dst = out + (size_t)tile * TILE_F4;
    #pragma unroll 8
    for (int i = 0; i < 32; ++i) {
        const int o4 = i * 256 + tid;    // linear v4f index in tile
        const int c4 = o4 & 15;
        const int m  = o4 >> 4;          // m = l*8 + n
        const int n  = m & 7;
        const int l  = m >> 3;
        const v4f v  = smem[n * BLK_F4 + l * 16 + c4];
        __builtin_nontemporal_store(v, &dst[o4]);  // TH=NT: don't thrash L2
    }
}

} // namespace

extern "C" void kernel_launch(void* const* d_in, const int* in_sizes, int n_in,
                              void* d_out, int out_size, void* d_ws, size_t ws_size,
                              hipStream_t stream)
{
    (void)n_in; (void)out_size; (void)d_ws; (void)ws_size;
    const float* E = (const float*)d_in[0];     // [B, H*W, L, C] fp32
    v4f* out = (v4f*)d_out;                     // [B, H*W, L*8, C] fp32

    // number of (b,p) tiles = total elements / (L*C)
    const int n_tiles = in_sizes[0] / (L * C);  // B * 256
    patch_neighbor_kernel<<<n_tiles, 256, 0, stream>>>(E, out);
}